// MoEAutoEncoder_28363964023538
// MI455X (gfx1250) — compile-verified
//
#include <hip/hip_runtime.h>
#include <cfloat>

// Problem constants (match reference)
#define B_N   4096
#define D_N   768
#define E_N   16
#define L_N   1536
#define KTOP  32
#define NT_PER_WAVE 12   // 96 n-tiles / 8 waves

typedef __attribute__((ext_vector_type(2))) float v2f;
typedef __attribute__((ext_vector_type(8))) float v8f;

__device__ __forceinline__ float waveSum(float v) {
#pragma unroll
  for (int off = 16; off > 0; off >>= 1) v += __shfl_xor(v, off, 32);
  return v;
}

// argmax with lowest-index tie-break (matches jax.lax.top_k ordering)
__device__ __forceinline__ void waveArgMax(float& v, int& i) {
#pragma unroll
  for (int off = 16; off > 0; off >>= 1) {
    float ov = __shfl_xor(v, off, 32);
    int   oi = __shfl_xor(i, off, 32);
    if (ov > v || (ov == v && oi < i)) { v = ov; i = oi; }
  }
}

// ---------------- kernel 0: zero routing counters ----------------
__global__ void init_counts_kernel(int* counts) {
  if (threadIdx.x < E_N) counts[threadIdx.x] = 0;
}

// ---------------- kernel 1: gating, top-2 routing, out = b_dec ----------------
// one wave per batch row; 8 waves per block
__global__ __launch_bounds__(256) void gate_kernel(
    const float* __restrict__ x, const float* __restrict__ Wg,
    const float* __restrict__ bg, const float* __restrict__ bgate,
    const float* __restrict__ bdec, float* __restrict__ out,
    int* __restrict__ counts, int* __restrict__ rlist, float* __restrict__ wlist) {
  const int wave = threadIdx.x >> 5, lane = threadIdx.x & 31;
  const int b = blockIdx.x * 8 + wave;

  // load x row (x - b_gate), 24 elems per lane
  float xr[24];
  const float* xp = x + (size_t)b * D_N;
#pragma unroll
  for (int j = 0; j < 24; ++j) xr[j] = xp[lane + 32 * j] - bgate[lane + 32 * j];

  // probs = relu((x - b_gate) @ Wg^T + bg); lane e (<16) keeps probs[e]
  float myp = 0.f;
  for (int e = 0; e < E_N; ++e) {
    const float* wp = Wg + (size_t)e * D_N;
    float s = 0.f;
#pragma unroll
    for (int j = 0; j < 24; ++j) s += xr[j] * wp[lane + 32 * j];
    s = waveSum(s);
    s = fmaxf(s + bg[e], 0.f);
    if (lane == e) myp = s;
  }

  // top-2 experts
  float v = (lane < E_N) ? myp : -FLT_MAX;
  float m1 = v; int i1 = lane; waveArgMax(m1, i1);
  float v2 = (lane == i1) ? -FLT_MAX : v;
  float m2 = v2; int i2 = lane; waveArgMax(m2, i2);

  // reference quirk: w[:, s] = probs[:, s] * mask[:, s] for s in {0,1}
  float p0 = __shfl(myp, 0, 32);
  float p1 = __shfl(myp, 1, 32);
  float w0 = (i1 == 0 || i2 == 0) ? p0 : 0.f;  // slot 0 -> expert i1, weight w0
  float w1 = (i1 == 1 || i2 == 1) ? p1 : 0.f;  // slot 1 -> expert i2, weight w1

  if (lane == 0) {
    if (w0 != 0.f) {
      int pos = atomicAdd(&counts[i1], 1);
      rlist[i1 * B_N + pos] = b; wlist[i1 * B_N + pos] = w0;
    }
    if (w1 != 0.f) {
      int pos = atomicAdd(&counts[i2], 1);
      rlist[i2 * B_N + pos] = b; wlist[i2 * B_N + pos] = w1;
    }
  }

  // out = b_dec (base; expert kernel atomically adds contributions)
  float* op = out + (size_t)b * D_N;
#pragma unroll
  for (int j = 0; j < 24; ++j) op[lane + 32 * j] = bdec[lane + 32 * j];
}

// ---------------- kernel 2: per-expert encode GEMM (WMMA f32) + top-32 + decode ----------------
// grid (tiles=256, experts=16), 256 threads (8 waves). Dynamic LDS:
//   Xs[16][768] gathered inputs, Zs[16][1536] activations, topv/topi[16][32]
__global__ __launch_bounds__(256) void expert_kernel(
    const float* __restrict__ x, const float* __restrict__ Wenc,
    const float* __restrict__ benc, const float* __restrict__ Wdec,
    const float* __restrict__ bdec, const int* __restrict__ counts,
    const int* __restrict__ rlist, const float* __restrict__ wlist,
    float* __restrict__ out) {
  const int e = blockIdx.y;
  const int tile = blockIdx.x;
  const int cnt = counts[e];
  if (tile * 16 >= cnt) return;
  const int nrows = min(16, cnt - tile * 16);

  extern __shared__ float smem[];
  float* Xs   = smem;                        // 16*768
  float* Zs   = Xs + 16 * D_N;               // 16*1536
  float* topv = Zs + 16 * L_N;               // 16*32
  int*   topi = (int*)(topv + 16 * KTOP);    // 16*32
  __shared__ int   s_row[16];
  __shared__ float s_w[16];

  const int tid = threadIdx.x;
  const int wave = tid >> 5, lane = tid & 31;
  const int half = lane >> 4, m16 = lane & 15;

  if (tid < 16) {
    if (tid < nrows) {
      s_row[tid] = rlist[e * B_N + tile * 16 + tid];
      s_w[tid]   = wlist[e * B_N + tile * 16 + tid];
    } else {
      s_row[tid] = 0; s_w[tid] = 0.f;
    }
  }
  __syncthreads();

  // stage gathered (x - b_dec) rows into LDS; pad rows with zeros
  for (int idx = tid; idx < 16 * D_N; idx += 256) {
    int r = idx / D_N, d = idx - r * D_N;
    Xs[idx] = (r < nrows) ? (x[(size_t)s_row[r] * D_N + d] - bdec[d]) : 0.f;
  }
  __syncthreads();

  // ---- GEMM: Z(16x1536) = Xs(16x768) * Wenc[e]^T, fp32 WMMA 16x16x4 ----
  const float* wep = Wenc + (size_t)e * L_N * D_N;
  v8f acc[NT_PER_WAVE];
#pragma unroll
  for (int t = 0; t < NT_PER_WAVE; ++t) acc[t] = (v8f)(0.f);

  for (int k = 0; k < D_N; k += 4) {
    const int ka = k + 2 * half;               // lanes 0-15: K=k..k+1, lanes 16-31: K=k+2..k+3
    v2f a;
    a.x = Xs[m16 * D_N + ka];
    a.y = Xs[m16 * D_N + ka + 1];
#pragma unroll
    for (int t = 0; t < NT_PER_WAVE; ++t) {
      const int nb = (t * 8 + wave) * 16;      // n-tile base
      const float* bp = wep + (size_t)(nb + m16) * D_N + ka;
      v2f bm; bm.x = bp[0]; bm.y = bp[1];      // B[k][n] = Wenc[e][n][k]
      acc[t] = __builtin_amdgcn_wmma_f32_16x16x4_f32(
          false, a, false, bm, (short)0, acc[t], false, false);
    }
  }

  // writeback: bias + relu into Zs.  C layout: VGPR j -> M = 8*half + j, N = m16
#pragma unroll
  for (int t = 0; t < NT_PER_WAVE; ++t) {
    const int n = (t * 8 + wave) * 16 + m16;
    const float be = benc[e * L_N + n];
#pragma unroll
    for (int j = 0; j < 8; ++j) {
      const int m = half * 8 + j;
      Zs[m * L_N + n] = fmaxf(acc[t][j] + be, 0.f);
    }
  }
  __syncthreads();

  // ---- top-32 per row (register-resident values, wave argmax per pass) ----
  for (int r = wave; r < 16; r += 8) {
    if (r >= nrows) continue;                  // wave-uniform
    float lv[48];
#pragma unroll
    for (int t = 0; t < 48; ++t) lv[t] = Zs[r * L_N + lane + 32 * t];
    float selv = 0.f; int seli = 0;
    for (int sel = 0; sel < KTOP; ++sel) {
      float bv = -1.f; int bj = 1 << 30;       // z >= 0 after relu
#pragma unroll
      for (int t = 0; t < 48; ++t) {
        if (lv[t] > bv) { bv = lv[t]; bj = lane + 32 * t; }  // strict > keeps lowest index
      }
      waveArgMax(bv, bj);
      if ((bj & 31) == lane) {                 // owning lane invalidates winner
        const int tw = bj >> 5;
#pragma unroll
        for (int tt = 0; tt < 48; ++tt) if (tt == tw) lv[tt] = -2.f;
      }
      if (lane == sel) { selv = bv; seli = bj; }
    }
    topv[r * KTOP + lane] = selv;
    topi[r * KTOP + lane] = seli;
  }
  __syncthreads();

  // ---- decode: out[b] += w * sum_k act_k * Wdec[e, idx_k, :] ----
  const float* wdp = Wdec + (size_t)e * L_N * D_N;
  for (int r = wave; r < 16; r += 8) {
    if (r >= nrows) continue;
    const int b = s_row[r];
    const float wgt = s_w[r];
    float dacc[24];
#pragma unroll
    for (int j = 0; j < 24; ++j) dacc[j] = 0.f;
    for (int kk = 0; kk < KTOP; ++kk) {
      const float vv = topv[r * KTOP + kk];
      const float* wr = wdp + (size_t)topi[r * KTOP + kk] * D_N;
#pragma unroll
      for (int j = 0; j < 24; ++j) dacc[j] += vv * wr[lane + 32 * j];
    }
    float* op = out + (size_t)b * D_N;
#pragma unroll
    for (int j = 0; j < 24; ++j)
      unsafeAtomicAdd(&op[lane + 32 * j], wgt * dacc[j]);   // global_atomic_add_f32
  }
}

// ---------------- host launch ----------------
extern "C" void kernel_launch(void* const* d_in, const int* in_sizes, int n_in,
                              void* d_out, int out_size, void* d_ws, size_t ws_size,
                              hipStream_t stream) {
  const float* x     = (const float*)d_in[0];
  const float* Wenc  = (const float*)d_in[1];
  const float* benc  = (const float*)d_in[2];
  const float* Wdec  = (const float*)d_in[3];
  const float* Wg    = (const float*)d_in[4];
  const float* bg    = (const float*)d_in[5];
  const float* bdec  = (const float*)d_in[6];
  const float* bgate = (const float*)d_in[7];
  float* out = (float*)d_out;

  // workspace layout: counts[16] | rlist[16*4096] | wlist[16*4096]
  int*   counts = (int*)d_ws;
  int*   rlist  = counts + 16;
  float* wlist  = (float*)(rlist + E_N * B_N);

  const size_t smem = (size_t)(16 * D_N + 16 * L_N + 16 * KTOP) * 4 + 16 * KTOP * 4;
  (void)hipFuncSetAttribute((const void*)expert_kernel,
                            hipFuncAttributeMaxDynamicSharedMemorySize, (int)smem);

  init_counts_kernel<<<1, 32, 0, stream>>>(counts);
  gate_kernel<<<B_N / 8, 256, 0, stream>>>(x, Wg, bg, bgate, bdec, out,
                                           counts, rlist, wlist);
  dim3 grid(B_N / 16, E_N);
  expert_kernel<<<grid, 256, smem, stream>>>(x, Wenc, benc, Wdec, bdec,
                                             counts, rlist, wlist, out);
}